// ConvDeepSet_36618891166165
// MI455X (gfx1250) — compile-verified
//
#include <hip/hip_runtime.h>
#include <math.h>

// ---------------------------------------------------------------------------
// ConvDeepSet on MI455X (gfx1250): separable RBF interpolation as two chained
// GEMMs per (batch, channel) using V_WMMA_F32_16X16X4_F32.
//
// Roofline: 30 GFLOP vs 280 MB of HBM traffic (97 MB intermediate stays in the
// 192 MB L2) -> HBM time ~12 us; fp32 WMMA sits >10x under the compute
// roofline, so we keep full f32 precision. The real second-order limit is L2
// operand traffic, so each wave register-blocks a 48x64 (stage1) / 48x32
// (stage2) tile: 12 / 6 accumulators, ~1 vmem issue per WMMA, and 720 = 15*48
// means no M-edge handling (EXEC stays all-1s through every WMMA).
// ---------------------------------------------------------------------------

typedef float v2f __attribute__((ext_vector_type(2)));
typedef float v8f __attribute__((ext_vector_type(8)));

constexpr int Bb = 8;     // batch
constexpr int Cc = 32;    // input channels
constexpr int Cp = 33;    // +density channel
constexpr int Wd = 256;   // lon input grid  (stage-1 K)
constexpr int Hd = 128;   // lat input grid  (stage-2 K)
constexpr int Xd = 720;   // lon output grid (= 15 tiles of 48)
constexpr int Yd = 361;   // lat output grid
constexpr int YP = 384;   // Y padded to 12 tiles of 32 (pad cols = 0)

// --------------------------- RBF weight kernels ----------------------------
// w0t[b][x][w] = exp(-0.5*(lon_in[b,w]-lon_out[b,x])^2 / ls^2)  (A: M=X, K=W)
__global__ void rbf_lon_kernel(const float* __restrict__ xin,
                               const float* __restrict__ xout,
                               const float* __restrict__ ls,
                               float* __restrict__ w0t) {
  int idx = blockIdx.x * blockDim.x + threadIdx.x;
  if (idx >= Bb * Xd * Wd) return;
  int w = idx % Wd;
  int t = idx / Wd;
  int x = t % Xd;
  int b = t / Xd;
  float l = ls[0];
  float inv = 0.5f / (l * l);
  float d = xin[b * Wd + w] - xout[b * Xd + x];
  w0t[idx] = __expf(-inv * d * d);
}

// w1p[b][h][y] = exp(-0.5*(lat_in[b,h]-lat_out[b,y])^2 / ls^2), zero-padded
__global__ void rbf_lat_kernel(const float* __restrict__ xin,
                               const float* __restrict__ xout,
                               const float* __restrict__ ls,
                               float* __restrict__ w1p) {
  int idx = blockIdx.x * blockDim.x + threadIdx.x;
  if (idx >= Bb * Hd * YP) return;
  int y = idx % YP;
  int t = idx / YP;
  int h = t % Hd;
  int b = t / Hd;
  float v = 0.0f;
  if (y < Yd) {
    float l = ls[0];
    float inv = 0.5f / (l * l);
    float d = xin[b * Hd + h] - xout[b * Yd + y];
    v = __expf(-inv * d * d);
  }
  w1p[idx] = v;
}

// ------------------------------ Stage 1 GEMM -------------------------------
// tmp[b,c][x][h] = sum_w w0t[b][x][w] * wt2[b,c][w][h]
// One wave per 48x64 tile (3 M-tiles x 4 N-tiles, 12 WMMA accumulators).
// NaN handling + density channel fused into the B-operand load.
__global__ void __launch_bounds__(32)
stage1_kernel(const float* __restrict__ wt,
              const float* __restrict__ w0t,
              float* __restrict__ tmp) {
  const int lane = threadIdx.x;            // 0..31
  const int m0 = blockIdx.x * 48;          // X tile base
  const int n0 = blockIdx.y * 64;          // H tile base
  const int bc = blockIdx.z;               // b*33 + c
  const int b = bc / Cp;
  const int c = bc % Cp;

  const int row  = lane & 15;              // A-matrix M within 16-tile
  const int half = lane >> 4;              // 0: K,K+1   1: K+2,K+3
  const int kh   = half * 2;
  const int col  = lane & 15;              // B/D-matrix N within 16-tile

  const float* A0 = w0t + (size_t)(b * Xd + m0 + row) * Wd;
  const float* Wt = wt + (size_t)(b * Cc + (c == 0 ? 0 : c - 1)) * Wd * Hd
                       + (n0 + col);
  const bool isDens = (c == 0);

  v8f acc[3][4];
#pragma unroll
  for (int mi = 0; mi < 3; ++mi)
#pragma unroll
    for (int ni = 0; ni < 4; ++ni) acc[mi][ni] = (v8f){};

  for (int k = 0; k < Wd; k += 4) {
    const int k0 = k + kh;
    v2f a[3];
#pragma unroll
    for (int mi = 0; mi < 3; ++mi)
      a[mi] = *(const v2f*)(A0 + (size_t)mi * 16 * Wd + k0);  // 8B, even offs
    v2f bb[4];
#pragma unroll
    for (int ni = 0; ni < 4; ++ni) {
      float r0 = Wt[(size_t)k0 * Hd + ni * 16];
      float r1 = Wt[(size_t)(k0 + 1) * Hd + ni * 16];
      if (isDens) {                                  // density mask channel
        bb[ni].x = (r0 == r0) ? 1.0f : 0.0f;
        bb[ni].y = (r1 == r1) ? 1.0f : 0.0f;
      } else {                                       // NaN -> 0
        bb[ni].x = (r0 == r0) ? r0 : 0.0f;
        bb[ni].y = (r1 == r1) ? r1 : 0.0f;
      }
    }
#pragma unroll
    for (int mi = 0; mi < 3; ++mi)
#pragma unroll
      for (int ni = 0; ni < 4; ++ni)
        acc[mi][ni] = __builtin_amdgcn_wmma_f32_16x16x4_f32(
            false, a[mi], false, bb[ni], (short)0, acc[mi][ni], false, false);
  }

  float* T = tmp + (size_t)bc * Xd * Hd;
#pragma unroll
  for (int mi = 0; mi < 3; ++mi)
#pragma unroll
    for (int ni = 0; ni < 4; ++ni)
#pragma unroll
      for (int i = 0; i < 8; ++i) {
        const int m = m0 + mi * 16 + i + half * 8;   // D VGPR i -> rows i/i+8
        T[(size_t)m * Hd + n0 + ni * 16 + col] = acc[mi][ni][i];
      }
}

// ------------------------------ Stage 2 GEMM -------------------------------
// ee[b,c][x][y] = sum_h tmp[b,c][x][h] * w1p[b][h][y]
// One wave per 48x32 tile (3x2 = 6 WMMA accumulators).
// DIVIDE=false: density channel (c=0), stored raw.
// DIVIDE=true : channels 1..32, divided by clip(density) in the epilogue.
template <bool DIVIDE>
__global__ void __launch_bounds__(32)
stage2_kernel(const float* __restrict__ tmp,
              const float* __restrict__ w1p,
              float* __restrict__ out) {
  const int lane = threadIdx.x;
  const int m0 = blockIdx.x * 48;          // X tile base
  const int n0 = blockIdx.y * 32;          // Y tile base (padded domain)
  int b, c;
  if (DIVIDE) { b = blockIdx.z >> 5; c = 1 + (blockIdx.z & 31); }
  else        { b = blockIdx.z;      c = 0; }
  const int bc = b * Cp + c;

  const int row  = lane & 15;
  const int half = lane >> 4;
  const int kh   = half * 2;
  const int col  = lane & 15;

  const float* A0 = tmp + ((size_t)bc * Xd + m0 + row) * Hd;
  const float* Bm = w1p + (size_t)b * Hd * YP + (n0 + col);

  v8f acc[3][2];
#pragma unroll
  for (int mi = 0; mi < 3; ++mi)
#pragma unroll
    for (int ni = 0; ni < 2; ++ni) acc[mi][ni] = (v8f){};

  for (int k = 0; k < Hd; k += 4) {
    const int k0 = k + kh;
    v2f a[3];
#pragma unroll
    for (int mi = 0; mi < 3; ++mi)
      a[mi] = *(const v2f*)(A0 + (size_t)mi * 16 * Hd + k0);
    v2f bb[2];
#pragma unroll
    for (int ni = 0; ni < 2; ++ni) {
      bb[ni].x = Bm[(size_t)k0 * YP + ni * 16];      // padded cols are 0
      bb[ni].y = Bm[(size_t)(k0 + 1) * YP + ni * 16];
    }
#pragma unroll
    for (int mi = 0; mi < 3; ++mi)
#pragma unroll
      for (int ni = 0; ni < 2; ++ni)
        acc[mi][ni] = __builtin_amdgcn_wmma_f32_16x16x4_f32(
            false, a[mi], false, bb[ni], (short)0, acc[mi][ni], false, false);
  }

  float* O = out + (size_t)bc * Xd * Yd;
  const float* Dns = out + (size_t)(b * Cp) * Xd * Yd;   // density channel
#pragma unroll
  for (int ni = 0; ni < 2; ++ni) {
    const int yy = n0 + ni * 16 + col;
    if (yy < Yd) {                           // per-lane guard, after all WMMAs
#pragma unroll
      for (int mi = 0; mi < 3; ++mi)
#pragma unroll
        for (int i = 0; i < 8; ++i) {
          const int m = m0 + mi * 16 + i + half * 8;
          float v = acc[mi][ni][i];
          if (DIVIDE) {
            float d = Dns[(size_t)m * Yd + yy];
            d = fminf(fmaxf(d, 1e-6f), 1e5f);
            v = v / d;
          }
          O[(size_t)m * Yd + yy] = v;
        }
    }
  }
}

// ------------------------------- launcher ----------------------------------
extern "C" void kernel_launch(void* const* d_in, const int* in_sizes, int n_in,
                              void* d_out, int out_size, void* d_ws, size_t ws_size,
                              hipStream_t stream) {
  (void)in_sizes; (void)n_in; (void)out_size; (void)ws_size;
  const float* wt   = (const float*)d_in[0];   // (B,C,W,H)
  const float* xil  = (const float*)d_in[1];   // (B,W)
  const float* xila = (const float*)d_in[2];   // (B,H)
  const float* xol  = (const float*)d_in[3];   // (B,X)
  const float* xola = (const float*)d_in[4];   // (B,Y)
  const float* ls   = (const float*)d_in[5];   // (1,)
  float* out = (float*)d_out;                  // (B,33,X,Y)

  char* ws = (char*)d_ws;
  float* w0t = (float*)ws;                                              // B*X*W
  float* w1p = (float*)(ws + (size_t)Bb * Xd * Wd * 4);                 // B*H*YP
  float* tmp = (float*)(ws + (size_t)Bb * Xd * Wd * 4
                           + (size_t)Bb * Hd * YP * 4);                 // B*33*X*H

  { int n = Bb * Xd * Wd;
    rbf_lon_kernel<<<(n + 255) / 256, 256, 0, stream>>>(xil, xol, ls, w0t); }
  { int n = Bb * Hd * YP;
    rbf_lat_kernel<<<(n + 255) / 256, 256, 0, stream>>>(xila, xola, ls, w1p); }

  // 15 x-tiles of 48; stage1 N = 128 = 2 tiles of 64
  stage1_kernel<<<dim3(Xd / 48, Hd / 64, Bb * Cp), dim3(32), 0, stream>>>(wt, w0t, tmp);

  // density channel first, then the divided channels (stream order = dep order)
  stage2_kernel<false><<<dim3(Xd / 48, YP / 32, Bb),      dim3(32), 0, stream>>>(tmp, w1p, out);
  stage2_kernel<true ><<<dim3(Xd / 48, YP / 32, Bb * Cc), dim3(32), 0, stream>>>(tmp, w1p, out);
}